// GA_2568390443642
// MI455X (gfx1250) — compile-verified
//
#include <hip/hip_runtime.h>
#include <hip/hip_bf16.h>
#include <math.h>

// Problem constants (match reference)
#define BB 4
#define CC 1024
#define TT 2048
#define C3 3072          // 3*CC
#define KKTOP 256        // TT / FACTOR

typedef float v2f __attribute__((ext_vector_type(2)));
typedef float v8f __attribute__((ext_vector_type(8)));

// ---------------------------------------------------------------------------
// CDNA5 async global->LDS copy (ASYNCcnt-tracked, ISA §15.18 op 98).
// Per lane: copies 16 bytes from global address g to the LDS offset taken from l.  Generic __shared__ pointers
// carry the wave-relative LDS offset in their low 32 bits (ISA §10.2).
// ---------------------------------------------------------------------------
__device__ __forceinline__ void async_copy_b128(const float* g, const float* l)
{
    unsigned lofs = (unsigned)(size_t)l;
    asm volatile("global_load_async_to_lds_b128 %0, %1, off"
                 :: "v"(lofs), "v"(g) : "memory");
}
__device__ __forceinline__ void wait_async0()
{
    asm volatile("s_wait_asynccnt 0x0" ::: "memory");
}

// ---------------------------------------------------------------------------
// fp32 WMMA GEMM:  C[M,N] = A[M,K] * B[K,N]  (+ optional Add[M,N])
//   TA: A[m][k] = Ag[k*lda + m]     TB: B[k][n] = Bg[n*ldb + k]
// Block: 256 threads = 8 waves; wave tile 32x32 (4 accumulators); block tile
// 64(M) x 128(N); K staged in 16-wide panels, double-buffered in LDS via
// GLOBAL_LOAD_ASYNC_TO_LDS_B128.  Compute: V_WMMA_F32_16X16X4_F32.
// LDS layouts keep each 16B global chunk contiguous in LDS; 20-float rows
// (80B) stay 16B-aligned and map strided reads onto 16 distinct banks.
// ---------------------------------------------------------------------------
template <bool TA, bool TB>
__global__ void gemm_wmma_f32(const float* __restrict__ Ag, long lda, long sA,
                              const float* __restrict__ Bg, long ldb, long sB,
                              float* __restrict__ Cg, long ldc, long sC,
                              int M, int N, int K,
                              const float* __restrict__ Addg, long ldadd, long sAdd)
{
    const int bz = blockIdx.z;
    Ag += (size_t)bz * sA;
    Bg += (size_t)bz * sB;
    Cg += (size_t)bz * sC;
    if (Addg) Addg += (size_t)bz * sAdd;

    const int n0 = blockIdx.x * 128;
    const int m0 = blockIdx.y * 64;

    const int tid  = threadIdx.x;
    const int wave = tid >> 5;
    const int lane = tid & 31;
    const int rc   = lane & 15;      // row (A/M) and col (B/N) within 16-tile
    const int kh   = lane >> 4;      // lane half selects K pair {0,1} vs {2,3}
    const int wmB  = (wave >> 2) * 32;   // wave M offset in block (0/32)
    const int wnB  = (wave & 3) * 32;    // wave N offset in block (0..96)

    // Double-buffered panels. TA: A as [k][64]; !TA: A as [m][20] (16 k + pad).
    //                         TB: B as [n][20];  !TB: B as [k][128].
    __shared__ __align__(16) float Abuf[2][TA ? (16 * 64) : (64 * 20)];
    __shared__ __align__(16) float Bbuf[2][TB ? (128 * 20) : (16 * 128)];

    auto stageA = [&](int buf, int kc) {
        const int e = tid;                       // 256 x 16B chunks = 64x16
        if (TA) {
            const int k = e >> 4, m4 = (e & 15) * 4;
            async_copy_b128(Ag + (size_t)(kc + k) * lda + (m0 + m4),
                            &Abuf[buf][k * 64 + m4]);
        } else {
            const int m = e >> 2, k4 = (e & 3) * 4;
            async_copy_b128(Ag + (size_t)(m0 + m) * lda + (kc + k4),
                            &Abuf[buf][m * 20 + k4]);
        }
    };
    auto stageB = [&](int buf, int kc) {
#pragma unroll
        for (int i = 0; i < 2; ++i) {
            const int g = tid + i * 256;         // 512 x 16B chunks = 16x128
            if (TB) {
                const int n = g >> 2, k4 = (g & 3) * 4;
                async_copy_b128(Bg + (size_t)(n0 + n) * ldb + (kc + k4),
                                &Bbuf[buf][n * 20 + k4]);
            } else {
                const int k = g >> 5, n4 = (g & 31) * 4;
                async_copy_b128(Bg + (size_t)(kc + k) * ldb + (n0 + n4),
                                &Bbuf[buf][k * 128 + n4]);
            }
        }
    };

    v8f acc[2][2] = {};   // [mTile][nTile]

    stageA(0, 0);
    stageB(0, 0);

    for (int kc = 0; kc < K; kc += 16) {
        const int buf = (kc >> 4) & 1;
        wait_async0();
        __syncthreads();
        if (kc + 16 < K) {
            stageA(buf ^ 1, kc + 16);
            stageB(buf ^ 1, kc + 16);
        }
        const float* Ab = Abuf[buf];
        const float* Bb = Bbuf[buf];

#pragma unroll
        for (int s = 0; s < 4; ++s) {
            const int kk0 = 4 * s + 2 * kh;
            v2f a0, a1, b0, b1;
            if (TA) {
                a0.x = Ab[(kk0) * 64 + wmB + rc];
                a0.y = Ab[(kk0 + 1) * 64 + wmB + rc];
                a1.x = Ab[(kk0) * 64 + wmB + 16 + rc];
                a1.y = Ab[(kk0 + 1) * 64 + wmB + 16 + rc];
            } else {
                a0.x = Ab[(wmB + rc) * 20 + kk0];
                a0.y = Ab[(wmB + rc) * 20 + kk0 + 1];
                a1.x = Ab[(wmB + 16 + rc) * 20 + kk0];
                a1.y = Ab[(wmB + 16 + rc) * 20 + kk0 + 1];
            }
            if (TB) {
                b0.x = Bb[(wnB + rc) * 20 + kk0];
                b0.y = Bb[(wnB + rc) * 20 + kk0 + 1];
                b1.x = Bb[(wnB + 16 + rc) * 20 + kk0];
                b1.y = Bb[(wnB + 16 + rc) * 20 + kk0 + 1];
            } else {
                b0.x = Bb[(kk0) * 128 + wnB + rc];
                b0.y = Bb[(kk0 + 1) * 128 + wnB + rc];
                b1.x = Bb[(kk0) * 128 + wnB + 16 + rc];
                b1.y = Bb[(kk0 + 1) * 128 + wnB + 16 + rc];
            }
            acc[0][0] = __builtin_amdgcn_wmma_f32_16x16x4_f32(
                false, a0, false, b0, (short)0, acc[0][0], false, false);
            acc[0][1] = __builtin_amdgcn_wmma_f32_16x16x4_f32(
                false, a0, false, b1, (short)0, acc[0][1], false, false);
            acc[1][0] = __builtin_amdgcn_wmma_f32_16x16x4_f32(
                false, a1, false, b0, (short)0, acc[1][0], false, false);
            acc[1][1] = __builtin_amdgcn_wmma_f32_16x16x4_f32(
                false, a1, false, b1, (short)0, acc[1][1], false, false);
        }
        // next iteration's wait_async + barrier orders buffer reuse
    }

    // D layout: VGPR r -> M = r + 8*laneHalf, N = lane&15
#pragma unroll
    for (int i = 0; i < 2; ++i) {
#pragma unroll
        for (int j = 0; j < 2; ++j) {
#pragma unroll
            for (int r = 0; r < 8; ++r) {
                const int m = m0 + wmB + i * 16 + r + 8 * kh;
                const int n = n0 + wnB + j * 16 + rc;
                float v = acc[i][j][r];
                if (Addg) v += Addg[(size_t)m * ldadd + n];
                Cg[(size_t)m * ldc + n] = v;
            }
        }
    }
}

// ---------------------------------------------------------------------------
// Distance-graph row constant: dc[i] = 1 / sum_{j != i} 1/|i-j|
// (the reference's renormalizations collapse to diff[i,j] = dc[i]/|i-j|)
// ---------------------------------------------------------------------------
__global__ void diffconst_kernel(float* __restrict__ dc)
{
    const int i = blockIdx.x;
    __shared__ float sh;
    if (threadIdx.x == 0) sh = 0.f;
    __syncthreads();
    float h = 0.f;
    for (int j = threadIdx.x; j < TT; j += blockDim.x) {
        int d = j - i; d = d < 0 ? -d : d;
        if (d != 0) h += 1.0f / (float)d;
    }
    atomicAdd(&sh, h);
    __syncthreads();
    if (threadIdx.x == 0) dc[i] = 1.0f / sh;
}

// ---------------------------------------------------------------------------
// Depthwise conv k=3, pad=1 (cross-correlation, matching lax.conv):
//   y[b,o,t] = w[o,0]*x[t-1] + w[o,1]*x[t] + w[o,2]*x[t+1]
// ---------------------------------------------------------------------------
__global__ void dwconv_kernel(const float* __restrict__ in,
                              const float* __restrict__ w_dw,
                              float* __restrict__ out)
{
    const size_t idx = (size_t)blockIdx.x * blockDim.x + threadIdx.x;
    const int t = (int)(idx % TT);
    const int o = (int)((idx / TT) % C3);
    const float xm = (t > 0)      ? in[idx - 1] : 0.f;
    const float xc = in[idx];
    const float xp = (t < TT - 1) ? in[idx + 1] : 0.f;
    out[idx] = w_dw[o * 3 + 0] * xm + w_dw[o * 3 + 1] * xc + w_dw[o * 3 + 2] * xp;
}

// ---------------------------------------------------------------------------
// L2 norms over channel axis for q (ch [0,C)) and k (ch [C,2C)), per (b,t).
// Grid: (TT/64, BB); block 256 = 4 channel-strips x 64 t's (coalesced).
// ---------------------------------------------------------------------------
__global__ void norm_reduce_kernel(const float* __restrict__ qkvn,
                                   float* __restrict__ invq,
                                   float* __restrict__ invk)
{
    const int b  = blockIdx.y;
    const int tc = threadIdx.x & 63;
    const int cr = threadIdx.x >> 6;
    const int t  = blockIdx.x * 64 + tc;
    const float* base = qkvn + (size_t)b * C3 * TT;

    float aq = 0.f, ak = 0.f;
    for (int c = cr; c < CC; c += 4) {
        float q = base[(size_t)c * TT + t];
        float k = base[(size_t)(CC + c) * TT + t];
        aq += q * q;
        ak += k * k;
    }
    __shared__ float rq[4][64];
    __shared__ float rk[4][64];
    rq[cr][tc] = aq;
    rk[cr][tc] = ak;
    __syncthreads();
    if (cr == 0) {
        float sq = rq[0][tc] + rq[1][tc] + rq[2][tc] + rq[3][tc];
        float sk = rk[0][tc] + rk[1][tc] + rk[2][tc] + rk[3][tc];
        invq[(size_t)b * TT + t] = 1.0f / fmaxf(sqrtf(sq), 1e-12f);
        invk[(size_t)b * TT + t] = 1.0f / fmaxf(sqrtf(sk), 1e-12f);
    }
}

__global__ void scale_kernel(float* __restrict__ qkvn,
                             const float* __restrict__ invq,
                             const float* __restrict__ invk)
{
    const size_t idx = (size_t)blockIdx.x * blockDim.x + threadIdx.x; // over B*2C*T
    const int t = (int)(idx % TT);
    const int c = (int)((idx / TT) % (2 * CC));
    const int b = (int)(idx / ((size_t)2 * CC * TT));
    const float inv = (c < CC) ? invq[(size_t)b * TT + t] : invk[(size_t)b * TT + t];
    qkvn[((size_t)b * C3 + c) * TT + t] *= inv;
}

// ---------------------------------------------------------------------------
// Per-row: sim row-normalize (with zero-sum guard), blend with distance graph,
// zero diagonal.  One block per (b,t) row.
// ---------------------------------------------------------------------------
__global__ void rownorm_diff_kernel(float* __restrict__ graph,
                                    const float* __restrict__ dc)
{
    const long rowid = blockIdx.x;           // b*T + t
    const int t = (int)(rowid % TT);
    float* row = graph + (size_t)rowid * TT;

    __shared__ float ssum;
    if (threadIdx.x == 0) ssum = 0.f;
    __syncthreads();
    float p = 0.f;
    for (int s = threadIdx.x; s < TT; s += 256) p += row[s];
    atomicAdd(&ssum, p);
    __syncthreads();
    const float rs  = ssum;
    const float inv = (rs == 0.f) ? 1.f : 1.f / rs;
    const float dct = dc[t];

    for (int s = threadIdx.x; s < TT; s += 256) {
        int d = s - t;
        float g;
        if (d == 0) {
            g = 0.f;
        } else {
            int ad = d < 0 ? -d : d;
            g = 0.5f * (row[s] * inv + dct / (float)ad);
        }
        row[s] = g;
    }
}

// ---------------------------------------------------------------------------
// Exact top-k (k=256) threshold per row via binary search on order-preserving
// float->uint keys, then renormalize surviving entries (zero-sum guard).
// One block per (b,t) row; row cached in LDS.
// ---------------------------------------------------------------------------
__device__ __forceinline__ unsigned fkey(float f)
{
    unsigned u = __float_as_uint(f);
    return u ^ ((u & 0x80000000u) ? 0xFFFFFFFFu : 0x80000000u);
}

__global__ void topk_norm_kernel(float* __restrict__ graph)
{
    const long rowid = blockIdx.x;
    float* row = graph + (size_t)rowid * TT;

    __shared__ float    rowv[TT];
    __shared__ unsigned keys[TT];
    __shared__ unsigned scnt;
    __shared__ float    ssum;

    for (int s = threadIdx.x; s < TT; s += 256) {
        float f = row[s];
        rowv[s] = f;
        keys[s] = fkey(f);
    }
    __syncthreads();

    // max x with |{key >= x}| >= KKTOP  ->  key of the k-th largest element
    unsigned lo = 0u, hi = 0xFFFFFFFFu;
    while (lo < hi) {
        unsigned mid = (unsigned)((((unsigned long long)lo + hi) + 1ull) >> 1);
        if (threadIdx.x == 0) scnt = 0u;
        __syncthreads();
        unsigned c = 0;
        for (int s = threadIdx.x; s < TT; s += 256) c += (keys[s] >= mid) ? 1u : 0u;
        atomicAdd(&scnt, c);
        __syncthreads();
        const unsigned total = scnt;
        __syncthreads();
        if (total >= KKTOP) lo = mid; else hi = mid - 1u;
    }
    const unsigned thr = lo;

    if (threadIdx.x == 0) ssum = 0.f;
    __syncthreads();
    float p = 0.f;
    for (int s = threadIdx.x; s < TT; s += 256)
        if (keys[s] >= thr) p += rowv[s];
    atomicAdd(&ssum, p);
    __syncthreads();
    const float rs  = ssum;
    const float inv = (rs == 0.f) ? 1.f : 1.f / rs;

    for (int s = threadIdx.x; s < TT; s += 256)
        row[s] = (keys[s] >= thr) ? rowv[s] * inv : 0.f;
}

// ---------------------------------------------------------------------------
// Launcher.  Workspace layout (floats):
//   [0)        qkv_raw : B*3C*T (96 MB) -- reused as graph (64 MB) later
//   [B*3C*T)   qkvn    : B*3C*T (96 MB)
//   [2*B*3C*T) dc:T, invq:B*T, invk:B*T
// Total ~202 MB.
// ---------------------------------------------------------------------------
extern "C" void kernel_launch(void* const* d_in, const int* in_sizes, int n_in,
                              void* d_out, int out_size, void* d_ws, size_t ws_size,
                              hipStream_t stream)
{
    const float* x     = (const float*)d_in[0];   // [B, C, T]
    const float* w_qkv = (const float*)d_in[1];   // [3C, C]
    const float* w_dw  = (const float*)d_in[2];   // [3C, 1, 3]
    float* out = (float*)d_out;                   // [B, C, T]

    const size_t nBig = (size_t)BB * C3 * TT;     // 25,165,824
    float* ws      = (float*)d_ws;
    float* qkv_raw = ws;
    float* qkvn    = ws + nBig;
    float* graph   = qkv_raw;                     // reuse after dwconv consumes raw
    float* dc      = ws + 2 * nBig;
    float* invq    = dc + TT;
    float* invk    = invq + (size_t)BB * TT;

    // 1) distance-graph row constants
    diffconst_kernel<<<TT, 256, 0, stream>>>(dc);

    // 2) qkv = W[3072x1024] * x_b[1024x2048]  (A normal, B normal)
    gemm_wmma_f32<false, false><<<dim3(TT / 128, C3 / 64, BB), 256, 0, stream>>>(
        w_qkv, (long)CC, 0L,
        x, (long)TT, (long)CC * TT,
        qkv_raw, (long)TT, (long)C3 * TT,
        C3, TT, CC,
        nullptr, 0L, 0L);

    // 3) depthwise conv k=3 pad=1
    dwconv_kernel<<<(unsigned)(nBig / 256), 256, 0, stream>>>(qkv_raw, w_dw, qkvn);

    // 4) channel L2 norms for q,k ; 5) scale in place
    norm_reduce_kernel<<<dim3(TT / 64, BB), 256, 0, stream>>>(qkvn, invq, invk);
    scale_kernel<<<(unsigned)(((size_t)BB * 2 * CC * TT) / 256), 256, 0, stream>>>(
        qkvn, invq, invk);

    // 6) sim[t,s] = sum_c qn[c,t]*kn[c,s]  (A transposed, B normal)
    gemm_wmma_f32<true, false><<<dim3(TT / 128, TT / 64, BB), 256, 0, stream>>>(
        qkvn, (long)TT, (long)C3 * TT,                       // q channels
        qkvn + (size_t)CC * TT, (long)TT, (long)C3 * TT,     // k channels
        graph, (long)TT, (long)TT * TT,
        TT, TT, CC,
        nullptr, 0L, 0L);

    // 7) row-normalize sim, blend distance graph, zero diagonal
    rownorm_diff_kernel<<<(unsigned)((size_t)BB * TT), 256, 0, stream>>>(graph, dc);

    // 8) top-k threshold + renormalize
    topk_norm_kernel<<<(unsigned)((size_t)BB * TT), 256, 0, stream>>>(graph);

    // 9) out[b,c,t] = v[b,c,t] + sum_s v[b,c,s]*graph[b,t,s]
    //    => D[CxT] = V[CxT] * graph^T  (A normal, B transposed) + V epilogue
    gemm_wmma_f32<false, true><<<dim3(TT / 128, CC / 64, BB), 256, 0, stream>>>(
        qkvn + (size_t)2 * CC * TT, (long)TT, (long)C3 * TT, // v channels
        graph, (long)TT, (long)TT * TT,
        out, (long)TT, (long)CC * TT,
        CC, TT, TT,
        qkvn + (size_t)2 * CC * TT, (long)TT, (long)C3 * TT);
}